// ClebschCombiningSingleUnrolled_51591147160307
// MI455X (gfx1250) — compile-verified
//
#include <hip/hip_runtime.h>

typedef __attribute__((ext_vector_type(2))) float v2f;
typedef __attribute__((ext_vector_type(4))) float v4f;
typedef __attribute__((ext_vector_type(8))) float v8f;

// Problem constants (fixed by the reference): M=9 m-channels, K=100 CG terms.
#define MCH 9
#define KTERMS 100
#define NGROUP (KTERMS / 4)   // 25 WMMA k-groups of 4
#define TCOLS 64              // columns staged per wave-tile (4 x 16-col WMMA sub-passes)
#define X2OFF (MCH * TCOLS * 4)   // byte offset of x2 slab within wave slab

// out(16x16) = A(16x4) x B(4x16) accumulated over 25 groups, f32 WMMA.
//   A[j, k]   = (mu[k]==j) ? mult[k] : 0      (wave-invariant, registers)
//   B[k, col] = X1[m1[k],col] * X2[m2[k],col] (gathered via per-wave LDS slab)
// Slab is staged 64 columns at a time with b128 global loads / ds stores;
// the 4 sub-tiles and the x2 slab are reached via ds immediate offsets, so the
// hoisted gather-address VGPRs are shared by all sub-passes.
__global__ __launch_bounds__(256) void cg_combine_wmma(
    const float* __restrict__ X1, const float* __restrict__ X2,
    const int* __restrict__ m1a, const int* __restrict__ m2a,
    const int* __restrict__ mua, const float* __restrict__ mult,
    float* __restrict__ out, int nt /* = N*D */)
{
    // Per-wave slab: x1 rows [0 .. 9*64), x2 rows after. 8 waves/block -> 36 KB.
    __shared__ float lds[8 * 2 * MCH * TCOLS];

    const int tid  = threadIdx.x;
    const int wave = tid >> 5;
    const int lane = tid & 31;
    const int half = lane >> 4;    // 0 = lanes 0-15, 1 = lanes 16-31
    const int l15  = lane & 15;    // A-row / B-column slot
    float* slab = &lds[wave * (2 * MCH * TCOLS)];

    // ---- Per-wave setup: A matrix (50 VGPRs) + packed half-resolved m-indices ----
    // A VGPR0 <- kslot (half?2:0), A VGPR1 <- kslot (half?3:1) per the 16x4 f32 A layout.
    float    A0[NGROUP], A1[NGROUP];
    unsigned packedM[NGROUP];      // bytes: {m1[ka], m2[ka], m1[kb], m2[kb]}, ka=4g+2*half
#pragma unroll
    for (int g = 0; g < NGROUP; ++g) {
        const int k0 = 4 * g;
        const int mu0 = mua[k0],  mu1 = mua[k0 + 1],  mu2 = mua[k0 + 2],  mu3 = mua[k0 + 3];
        const float w0 = mult[k0], w1 = mult[k0 + 1], w2 = mult[k0 + 2], w3 = mult[k0 + 3];
        const int   muA = half ? mu2 : mu0;
        const int   muB = half ? mu3 : mu1;
        const float wA  = half ? w2 : w0;
        const float wB  = half ? w3 : w1;
        A0[g] = (l15 == muA) ? wA : 0.0f;
        A1[g] = (l15 == muB) ? wB : 0.0f;
        const unsigned lo = (unsigned)m1a[k0]     | ((unsigned)m2a[k0]     << 8) |
                            ((unsigned)m1a[k0 + 1] << 16) | ((unsigned)m2a[k0 + 1] << 24);
        const unsigned hi = (unsigned)m1a[k0 + 2] | ((unsigned)m2a[k0 + 2] << 8) |
                            ((unsigned)m1a[k0 + 3] << 16) | ((unsigned)m2a[k0 + 3] << 24);
        packedM[g] = half ? hi : lo;
    }

    // Staging roles: low half streams X1 rows, high half streams X2 rows (float4).
    const float* Xsrc = half ? X2 : X1;
    float* slabDst = slab + half * (MCH * TCOLS) + 4 * l15;
    const char* slabB = (const char*)slab;                  // byte-addressed gather
    const unsigned gbase1 = (unsigned)(l15 << 2);           // x1 gather base (bytes)

    const int ntiles    = nt >> 6;                          // 64-col tiles
    const int waveCount = (gridDim.x * blockDim.x) >> 5;
    int tile = (blockIdx.x * (blockDim.x >> 5)) + wave;

    for (; tile < ntiles; tile += waveCount) {
        const int colbase = tile << 6;

        // Stage 9x64 of each input: 9 x b128 per half-wave (256B contiguous/row).
#pragma unroll
        for (int m = 0; m < MCH; ++m) {
            v4f t = *(const v4f*)(Xsrc + (size_t)m * (size_t)nt + (size_t)(colbase + 4 * l15));
            *(v4f*)(slabDst + m * TCOLS) = t;
        }
        // Same-wave DS ops are in-order: gather loads below see these stores.

#pragma unroll
        for (int sub = 0; sub < 4; ++sub) {
            const unsigned soff = (unsigned)(sub * 64);     // folds into ds offset imm
            v8f acc = {0.f, 0.f, 0.f, 0.f, 0.f, 0.f, 0.f, 0.f};
#pragma unroll
            for (int g = 0; g < NGROUP; ++g) {
                const unsigned pm = packedM[g];
                const float x1a = *(const float*)(slabB + (((pm      ) & 15u) << 8) + gbase1 + soff);
                const float x2a = *(const float*)(slabB + (((pm >>  8) & 15u) << 8) + gbase1 + soff + X2OFF);
                const float x1b = *(const float*)(slabB + (((pm >> 16) & 15u) << 8) + gbase1 + soff);
                const float x2b = *(const float*)(slabB + (((pm >> 24) & 15u) << 8) + gbase1 + soff + X2OFF);
                v2f a = {A0[g], A1[g]};
                v2f b = {x1a * x2a, x1b * x2b};
                acc = __builtin_amdgcn_wmma_f32_16x16x4_f32(
                    /*neg_a=*/false, a, /*neg_b=*/false, b,
                    /*c_mod=*/(short)0, acc, /*reuse_a=*/false, /*reuse_b=*/false);
            }
            // D layout: VGPR r = row r (lanes 0-15) / row r+8 (lanes 16-31).
            // Rows 9..15 are exactly zero (A is zero there) and are not stored.
            const int col = colbase + sub * 16 + l15;
#pragma unroll
            for (int r = 0; r < 8; ++r) {
                const int orow = r + (half << 3);
                if (orow < MCH)
                    out[(size_t)orow * (size_t)nt + (size_t)col] = acc[r];
            }
        }
    }
}

extern "C" void kernel_launch(void* const* d_in, const int* in_sizes, int n_in,
                              void* d_out, int out_size, void* d_ws, size_t ws_size,
                              hipStream_t stream) {
    const float* X1   = (const float*)d_in[0];
    const float* X2   = (const float*)d_in[1];
    const int*   m1a  = (const int*)d_in[2];
    const int*   m2a  = (const int*)d_in[3];
    const int*   mua  = (const int*)d_in[4];
    const float* mult = (const float*)d_in[5];
    float* out = (float*)d_out;

    const int nt = in_sizes[0] / MCH;   // N*D = 1,048,576

    dim3 grid(512), block(256);         // 4096 waves, 4 x 64-column tiles each
    cg_combine_wmma<<<grid, block, 0, stream>>>(X1, X2, m1a, m2a, mua, mult, out, nt);
}